// EfficientGlobalPointer_35304631173234
// MI455X (gfx1250) — compile-verified
//
#include <hip/hip_runtime.h>
#include <math.h>

typedef float v2f __attribute__((ext_vector_type(2)));
typedef float v8f __attribute__((ext_vector_type(8)));

#define HEAD_SZ   64
#define NHEADS    12
#define HDIM      768
#define NDIM      128          // 2*HEAD_SZ
#define BSZ       16
#define LSZ       512
#define BIAS_N    24           // 2*NHEADS
#define BIG_NEG   1000000000000.0f

static __device__ __forceinline__ v8f wmma_f32_4(v2f a, v2f b, v8f c) {
    // D = A(16x4,f32) * B(4x16,f32) + C(16x16,f32)
    return __builtin_amdgcn_wmma_f32_16x16x4_f32(
        /*neg_a=*/false, a, /*neg_b=*/false, b,
        /*c_mod=*/(short)0, c, /*reuse_a=*/false, /*reuse_b=*/false);
}

// ---------------------------------------------------------------------------
// Kernel 1: seq[M=8192, 128] = hidden[M,768] @ pW[768,128] + pb
// One wave per 16x16 output tile; K swept in steps of 4 via v_wmma_f32_16x16x4.
// ---------------------------------------------------------------------------
__global__ void __launch_bounds__(32)
seq_gemm_kernel(const float* __restrict__ hidden,
                const float* __restrict__ pW,
                const float* __restrict__ pb,
                float* __restrict__ seq) {
    const int m0   = blockIdx.x * 16;          // 512 tiles over M
    const int n0   = blockIdx.y * 16;          // 8 tiles over N
    const int lane = threadIdx.x;
    const int half = lane >> 4;                // 0: lanes 0-15, 1: lanes 16-31
    const int l16  = lane & 15;

    const float* Arow = hidden + (size_t)(m0 + l16) * HDIM;   // A: row m0+l16
    const int nB = n0 + l16;                                   // B: col n0+l16

    v8f acc = {};
    #pragma unroll 4
    for (int k = 0; k < HDIM; k += 4) {
        v2f a, b;
        // A 16x4: V0 = K(k+2*half), V1 = K(k+2*half+1)  -> contiguous pair
        a.x = Arow[k + 2 * half + 0];
        a.y = Arow[k + 2 * half + 1];
        // B 4x16: lanes0-15 -> K=k+0/k+1 ; lanes16-31 -> K=k+2/k+3
        b.x = pW[(size_t)(k + 2 * half + 0) * NDIM + nB];
        b.y = pW[(size_t)(k + 2 * half + 1) * NDIM + nB];
        acc = wmma_f32_4(a, b, acc);
    }

    const float bias = pb[nB];
    #pragma unroll
    for (int r = 0; r < 8; ++r) {
        const int m = m0 + 8 * half + r;
        seq[(size_t)m * NDIM + n0 + l16] = acc[r] + bias;
    }
}

// ---------------------------------------------------------------------------
// Kernel 2: biasRaw[row, j] = seq[row,:] . qW[:,j] + qb[j]   (row-major B*L x 24)
// Must run BEFORE rope (uses pre-rope seq).
// ---------------------------------------------------------------------------
__global__ void __launch_bounds__(256)
bias_gemm_kernel(const float* __restrict__ seq,
                 const float* __restrict__ qW,
                 const float* __restrict__ qb,
                 float* __restrict__ biasRaw) {
    const int idx = blockIdx.x * blockDim.x + threadIdx.x;   // B*L*24 threads
    const int row = idx / BIAS_N;
    const int j   = idx - row * BIAS_N;
    const float* s = seq + (size_t)row * NDIM;
    float acc = qb[j];
    #pragma unroll 8
    for (int k = 0; k < NDIM; ++k)
        acc = fmaf(s[k], qW[(size_t)k * BIAS_N + j], acc);
    biasRaw[(size_t)row * BIAS_N + j] = acc;
}

// ---------------------------------------------------------------------------
// Kernel 3: in-place RoPE on both 64-wide halves of each seq row.
// thread -> (row, segment q/k, pair i in 0..31)
// ---------------------------------------------------------------------------
__global__ void __launch_bounds__(256)
rope_kernel(float* __restrict__ seq) {
    const int idx = blockIdx.x * blockDim.x + threadIdx.x;   // B*L*64 threads
    const int row = idx >> 6;
    const int p   = idx & 63;
    const int seg = p >> 5;                 // 0 = q half, 1 = k half
    const int i   = p & 31;                 // rotary pair index
    const int pos = row & (LSZ - 1);        // position within sequence

    const float inv = expf((float)(2 * i) * (-logf(10000.0f) / (float)HEAD_SZ));
    const float ang = (float)pos * inv;
    const float c = cosf(ang), s = sinf(ang);

    float* ptr = seq + (size_t)row * NDIM + seg * HEAD_SZ + 2 * i;
    const float x0 = ptr[0], x1 = ptr[1];
    ptr[0] = x0 * c - x1 * s;
    ptr[1] = x1 * c + x0 * s;
}

// ---------------------------------------------------------------------------
// Kernel 4: per (b, m-tile, n-tile): S = qw @ kw^T / 8 via WMMA, then fan out
// to all 12 heads with bias/mask applied, streamed with non-temporal stores.
// ---------------------------------------------------------------------------
__global__ void __launch_bounds__(32)
logits_kernel(const float* __restrict__ seq,       // post-rope
              const float* __restrict__ biasRaw,   // (B*L, 24)
              const int*   __restrict__ amask,     // (B, L)
              float* __restrict__ out) {           // (B, 12, L, L)
    const int n0 = blockIdx.x * 16;
    const int m0 = blockIdx.y * 16;
    const int b  = blockIdx.z;
    const int lane = threadIdx.x;
    const int half = lane >> 4;
    const int l16  = lane & 15;

    const float* qrow = seq + ((size_t)b * LSZ + m0 + l16) * NDIM;            // q cols 0..63
    const float* krow = seq + ((size_t)b * LSZ + n0 + l16) * NDIM + HEAD_SZ;  // k cols 64..127

    v8f acc = {};
    #pragma unroll
    for (int k = 0; k < HEAD_SZ; k += 4) {
        v2f a, bb;
        a.x  = qrow[k + 2 * half + 0];
        a.y  = qrow[k + 2 * half + 1];
        // B[K][N] = kw[n0+N][K]; per-lane N fixed, K advances -> contiguous pair
        bb.x = krow[k + 2 * half + 0];
        bb.y = krow[k + 2 * half + 1];
        acc = wmma_f32_4(a, bb, acc);
    }

    const int   n    = n0 + l16;
    const float pm   = (float)amask[(size_t)b * LSZ + n];   // pad mask along n
    const float padd = -(1.0f - pm) * BIG_NEG;

    // scaled dot, shared by all heads
    float s[8];
    #pragma unroll
    for (int r = 0; r < 8; ++r) s[r] = acc[r] * 0.125f;

    const float* b0base = biasRaw + ((size_t)b * LSZ) * BIAS_N;   // bias0: index 2h at row m
    const float* b1row  = biasRaw + ((size_t)b * LSZ + n) * BIAS_N; // bias1: index 2h+1 at row n

    #pragma unroll 2
    for (int h = 0; h < NHEADS; ++h) {
        const float b1 = b1row[2 * h + 1];
        float* obase = out + (((size_t)b * NHEADS + h) * LSZ) * LSZ;
        #pragma unroll
        for (int r = 0; r < 8; ++r) {
            const int m = m0 + 8 * half + r;
            const float b0 = b0base[(size_t)m * BIAS_N + 2 * h];
            float v = s[r] + b0 + b1;
            v = v * pm + padd;                       // pad mask (exact ref order)
            if (n < m) v -= BIG_NEG;                 // strict-lower-tri causal mask
            __builtin_nontemporal_store(v, obase + (size_t)m * LSZ + n);
        }
    }
}

// ---------------------------------------------------------------------------
extern "C" void kernel_launch(void* const* d_in, const int* in_sizes, int n_in,
                              void* d_out, int out_size, void* d_ws, size_t ws_size,
                              hipStream_t stream) {
    (void)in_sizes; (void)n_in; (void)out_size; (void)ws_size;

    const float* hidden = (const float*)d_in[0];   // (16,512,768)
    const int*   amask  = (const int*)  d_in[1];   // (16,512)
    const float* pW     = (const float*)d_in[2];   // (768,128)
    const float* pb     = (const float*)d_in[3];   // (128,)
    const float* qW     = (const float*)d_in[4];   // (128,24)
    const float* qb     = (const float*)d_in[5];   // (24,)
    float*       out    = (float*)d_out;           // (16,12,512,512)

    float* seq     = (float*)d_ws;                               // 16*512*128 f32 = 4 MB
    float* biasRaw = seq + (size_t)BSZ * LSZ * NDIM;             // 16*512*24 f32 = 0.75 MB

    // 1) seq = hidden @ pW + pb   (WMMA f32 16x16x4)
    seq_gemm_kernel<<<dim3(BSZ * LSZ / 16, NDIM / 16), 32, 0, stream>>>(hidden, pW, pb, seq);

    // 2) biasRaw = seq @ qW + qb  (pre-rope seq)
    bias_gemm_kernel<<<(BSZ * LSZ * BIAS_N) / 256, 256, 0, stream>>>(seq, qW, qb, biasRaw);

    // 3) in-place RoPE on q/k halves of seq
    rope_kernel<<<(BSZ * LSZ * 64) / 256, 256, 0, stream>>>(seq);

    // 4) fused QK^T (WMMA) + bias + masks, NT-streamed 201 MB output
    logits_kernel<<<dim3(LSZ / 16, LSZ / 16, BSZ), 32, 0, stream>>>(seq, biasRaw, amask, out);
}